// GNNActor_54503134986926
// MI455X (gfx1250) — compile-verified
//
#include <hip/hip_runtime.h>

#define N_NODES 120000
#define N_EDGES 1200000
#define C_IN    64
#define HID     256
#define ROWS    32            // node rows per fused-MLP block

typedef float v2f __attribute__((ext_vector_type(2)));
typedef float v8f __attribute__((ext_vector_type(8)));

__device__ __forceinline__ float leaky01(float v) { return v >= 0.0f ? v : 0.01f * v; }

__device__ __forceinline__ void atomAddF(float* p, float v) {
    __hip_atomic_fetch_add(p, v, __ATOMIC_RELAXED, __HIP_MEMORY_SCOPE_AGENT);
}

__device__ __forceinline__ v8f wmma4(v2f a, v2f b, v8f c) {
    return __builtin_amdgcn_wmma_f32_16x16x4_f32(false, a, false, b, (short)0, c,
                                                 false, false);
}

// ---------------- degree / normalization ----------------
__global__ void deg_init_kernel(float* __restrict__ deg, int n) {
    int i = blockIdx.x * blockDim.x + threadIdx.x;
    if (i < n) deg[i] = 1.0f;                 // self loop contributes 1
}

__global__ void deg_accum_kernel(const int* __restrict__ ei, float* __restrict__ deg, int e) {
    int i = blockIdx.x * blockDim.x + threadIdx.x;
    if (i < e) atomAddF(&deg[ei[N_EDGES + i]], 1.0f);    // dst row of edge_index
}

__global__ void deg_rsqrt_kernel(float* __restrict__ deg, int n) {
    int i = blockIdx.x * blockDim.x + threadIdx.x;
    if (i < n) deg[i] = rsqrtf(deg[i]);       // deg >= 1 always (self loops)
}

// ---------------- h = x @ W_gcn  (N x 64) @ (64 x 64), f32 WMMA ----------------
// 256 threads = 8 waves. Block covers 64 rows x 64 cols = 16 tiles; each wave owns
// 2 row-tiles sharing the same B fragment (2 independent wmma chains).
__global__ __launch_bounds__(256) void gemm_xw_kernel(const float* __restrict__ x,
                                                      const float* __restrict__ W,
                                                      float* __restrict__ h, int nrows) {
    int tid  = threadIdx.x;
    int wave = tid >> 5, lane = tid & 31;
    int half = lane >> 4, lm = lane & 15;
    int row0 = blockIdx.x * 64 + (wave >> 2) * 16;   // and row0+32
    int colBase = (wave & 3) * 16;
    if (row0 >= nrows) return;

    v8f acc0 = {}, acc1 = {};
    // A (16x4 f32): lanes 0-15 hold K=0,1 in v0,v1; lanes 16-31 hold K=2,3
    const float* a0 = x + (size_t)(row0 + lm) * C_IN + 2 * half;
    const float* a1 = x + (size_t)(row0 + 32 + lm) * C_IN + 2 * half;
    #pragma unroll
    for (int k = 0; k < C_IN; k += 4) {
        v2f av0, av1, b;
        av0.x = a0[k];  av0.y = a0[k + 1];
        av1.x = a1[k];  av1.y = a1[k + 1];
        // B (4x16 f32): v0 = rows K, K+2 ; v1 = rows K+1, K+3 striped across lanes
        b.x = W[(size_t)(k + 2 * half) * C_IN + colBase + lm];
        b.y = W[(size_t)(k + 1 + 2 * half) * C_IN + colBase + lm];
        acc0 = wmma4(av0, b, acc0);
        acc1 = wmma4(av1, b, acc1);
    }
    // D: VGPR i -> row (i + 8*half), col (colBase + lm)
    #pragma unroll
    for (int i = 0; i < 8; ++i) {
        h[(size_t)(row0 + i + 8 * half) * C_IN + colBase + lm]      = acc0[i];
        h[(size_t)(row0 + 32 + i + 8 * half) * C_IN + colBase + lm] = acc1[i];
    }
}

// ---------------- agg init with self-loop message: agg = h * dinv^2 ----------------
__global__ void agg_init_kernel(const float* __restrict__ h, const float* __restrict__ dinv,
                                float* __restrict__ agg, int total) {
    int i = blockIdx.x * blockDim.x + threadIdx.x;
    if (i < total) {
        int n = i >> 6;                  // / C_IN
        float di = dinv[n];
        agg[i] = h[i] * di * di;
    }
}

// ---------------- edge scatter: agg[dst] += h[src] * dinv[src]*dinv[dst] ----------------
// One thread per (edge, 4 channels): float4 read, 4 native f32 atomics.
__global__ void scatter_kernel(const int* __restrict__ ei, const float* __restrict__ h,
                               const float* __restrict__ dinv, float* __restrict__ agg,
                               int total /* E * 16 */) {
    int i = blockIdx.x * blockDim.x + threadIdx.x;
    if (i < total) {
        int e  = i >> 4;
        int c4 = (i & 15) << 2;
        int s = ei[e];
        int d = ei[N_EDGES + e];
        float norm = dinv[s] * dinv[d];
        const float4 hv = *(const float4*)(h + (size_t)s * C_IN + c4);
        float* ap = agg + (size_t)d * C_IN + c4;
        atomAddF(ap + 0, hv.x * norm);
        atomAddF(ap + 1, hv.y * norm);
        atomAddF(ap + 2, hv.z * norm);
        atomAddF(ap + 3, hv.w * norm);
    }
}

// ---------------- fused MLP: z=relu(agg+b)+x ; h1=leaky(z W1^T+b1) ;
//                  h2=leaky(h1 W2^T+b2) ; out = h2 . Wd + bd ----------------
// 256 threads = 8 waves, ROWS=32 node rows per block. Each wave computes a 2x2
// grid of 16x16 tiles (4 independent wmma chains, A/B fragments each reused 2x).
__global__ __launch_bounds__(256) void mlp_fused_kernel(
    const float* __restrict__ agg, const float* __restrict__ x,
    const float* __restrict__ b_gcn, const float* __restrict__ W1,
    const float* __restrict__ b1, const float* __restrict__ W2,
    const float* __restrict__ b2, const float* __restrict__ Wd,
    const float* __restrict__ bd, float* __restrict__ out) {
    __shared__ float zb[ROWS * (C_IN + 1)];   // stride 65  -> conflict-free A gathers
    __shared__ float h1[ROWS * (HID + 1)];    // stride 257
    __shared__ float h2[ROWS * (HID + 1)];

    int tid = threadIdx.x;
    int nodeBase = blockIdx.x * ROWS;

    // stage 0: z = relu(agg + b_gcn) + x  (residual)
    for (int i = tid; i < ROWS * C_IN; i += 256) {
        int r = i >> 6, c = i & (C_IN - 1);
        size_t g = (size_t)(nodeBase + r) * C_IN + c;
        float v = agg[g] + b_gcn[c];
        v = v > 0.0f ? v : 0.0f;
        zb[r * (C_IN + 1) + c] = v + x[g];
    }
    __syncthreads();

    int wave = tid >> 5, lane = tid & 31, half = lane >> 4, lm = lane & 15;
    int col0 = wave * 16;
    int col1 = col0 + 128;

    // stage 1: h1[32][256] = leaky(z @ W1^T + b1)
    {
        v8f acc00 = {}, acc01 = {}, acc10 = {}, acc11 = {};
        const float* aR0 = zb + lm * (C_IN + 1) + 2 * half;
        const float* aR1 = zb + (16 + lm) * (C_IN + 1) + 2 * half;
        const float* w0  = W1 + (size_t)(col0 + lm) * C_IN + 2 * half; // B[k][j]=W1[j][k]
        const float* w1  = W1 + (size_t)(col1 + lm) * C_IN + 2 * half;
        #pragma unroll
        for (int k = 0; k < C_IN; k += 4) {
            v2f a0v, a1v, b0v, b1v;
            a0v.x = aR0[k]; a0v.y = aR0[k + 1];
            a1v.x = aR1[k]; a1v.y = aR1[k + 1];
            b0v.x = w0[k];  b0v.y = w0[k + 1];
            b1v.x = w1[k];  b1v.y = w1[k + 1];
            acc00 = wmma4(a0v, b0v, acc00);
            acc01 = wmma4(a0v, b1v, acc01);
            acc10 = wmma4(a1v, b0v, acc10);
            acc11 = wmma4(a1v, b1v, acc11);
        }
        float bc0 = b1[col0 + lm], bc1 = b1[col1 + lm];
        #pragma unroll
        for (int i = 0; i < 8; ++i) {
            int r = i + 8 * half;
            h1[r * (HID + 1) + col0 + lm]        = leaky01(acc00[i] + bc0);
            h1[r * (HID + 1) + col1 + lm]        = leaky01(acc01[i] + bc1);
            h1[(16 + r) * (HID + 1) + col0 + lm] = leaky01(acc10[i] + bc0);
            h1[(16 + r) * (HID + 1) + col1 + lm] = leaky01(acc11[i] + bc1);
        }
    }
    __syncthreads();

    // stage 2: h2[32][256] = leaky(h1 @ W2^T + b2)
    {
        v8f acc00 = {}, acc01 = {}, acc10 = {}, acc11 = {};
        const float* aR0 = h1 + lm * (HID + 1) + 2 * half;
        const float* aR1 = h1 + (16 + lm) * (HID + 1) + 2 * half;
        const float* w0  = W2 + (size_t)(col0 + lm) * HID + 2 * half;  // B[k][j]=W2[j][k]
        const float* w1  = W2 + (size_t)(col1 + lm) * HID + 2 * half;
        #pragma unroll 4
        for (int k = 0; k < HID; k += 4) {
            v2f a0v, a1v, b0v, b1v;
            a0v.x = aR0[k]; a0v.y = aR0[k + 1];
            a1v.x = aR1[k]; a1v.y = aR1[k + 1];
            b0v.x = w0[k];  b0v.y = w0[k + 1];
            b1v.x = w1[k];  b1v.y = w1[k + 1];
            acc00 = wmma4(a0v, b0v, acc00);
            acc01 = wmma4(a0v, b1v, acc01);
            acc10 = wmma4(a1v, b0v, acc10);
            acc11 = wmma4(a1v, b1v, acc11);
        }
        float bc0 = b2[col0 + lm], bc1 = b2[col1 + lm];
        #pragma unroll
        for (int i = 0; i < 8; ++i) {
            int r = i + 8 * half;
            h2[r * (HID + 1) + col0 + lm]        = leaky01(acc00[i] + bc0);
            h2[r * (HID + 1) + col1 + lm]        = leaky01(acc01[i] + bc1);
            h2[(16 + r) * (HID + 1) + col0 + lm] = leaky01(acc10[i] + bc0);
            h2[(16 + r) * (HID + 1) + col1 + lm] = leaky01(acc11[i] + bc1);
        }
    }
    __syncthreads();

    // stage 3: out[n] = h2[n,:] . Wd + bd   (8 partials of 32 elems per row)
    {
        int row = tid >> 3, seg = tid & 7;
        const float* hrow = h2 + row * (HID + 1) + seg * 32;
        const float* wseg = Wd + seg * 32;
        float p = 0.0f;
        #pragma unroll
        for (int c = 0; c < 32; ++c) p += hrow[c] * wseg[c];
        zb[row * 8 + seg] = p;             // reuse z buffer (>= 256 floats)
    }
    __syncthreads();
    if (tid < ROWS) {
        float acc = 0.0f;
        #pragma unroll
        for (int s = 0; s < 8; ++s) acc += zb[tid * 8 + s];
        out[nodeBase + tid] = acc + bd[0];
    }
}

extern "C" void kernel_launch(void* const* d_in, const int* in_sizes, int n_in,
                              void* d_out, int out_size, void* d_ws, size_t ws_size,
                              hipStream_t stream) {
    const float* x     = (const float*)d_in[0];
    const int*   ei    = (const int*)d_in[1];
    const float* W_gcn = (const float*)d_in[2];
    const float* b_gcn = (const float*)d_in[3];
    const float* W1    = (const float*)d_in[4];
    const float* b1    = (const float*)d_in[5];
    const float* W2    = (const float*)d_in[6];
    const float* b2    = (const float*)d_in[7];
    const float* Wd    = (const float*)d_in[8];
    const float* bd    = (const float*)d_in[9];
    float* out = (float*)d_out;

    // workspace: dinv[N] | h[N*C] | agg[N*C]  (~62 MB)
    char* ws = (char*)d_ws;
    size_t off0 = 0;
    size_t off1 = (off0 + (size_t)N_NODES * sizeof(float) + 255) & ~(size_t)255;
    size_t off2 = (off1 + (size_t)N_NODES * C_IN * sizeof(float) + 255) & ~(size_t)255;
    float* dinv = (float*)(ws + off0);
    float* h    = (float*)(ws + off1);
    float* agg  = (float*)(ws + off2);

    const int B = 256;
    deg_init_kernel<<<(N_NODES + B - 1) / B, B, 0, stream>>>(dinv, N_NODES);
    deg_accum_kernel<<<(N_EDGES + B - 1) / B, B, 0, stream>>>(ei, dinv, N_EDGES);
    deg_rsqrt_kernel<<<(N_NODES + B - 1) / B, B, 0, stream>>>(dinv, N_NODES);

    gemm_xw_kernel<<<N_NODES / 64, B, 0, stream>>>(x, W_gcn, h, N_NODES);

    agg_init_kernel<<<((size_t)N_NODES * C_IN + B - 1) / B, B, 0, stream>>>(
        h, dinv, agg, N_NODES * C_IN);
    scatter_kernel<<<((size_t)N_EDGES * 16 + B - 1) / B, B, 0, stream>>>(
        ei, h, dinv, agg, N_EDGES * 16);

    mlp_fused_kernel<<<N_NODES / ROWS, B, 0, stream>>>(agg, x, b_gcn, W1, b1, W2, b2,
                                                       Wd, bd, out);
}